// MxDNAConvMoeBlock_11690900980376
// MI455X (gfx1250) — compile-verified
//
#include <hip/hip_runtime.h>
#include <math.h>

typedef __attribute__((ext_vector_type(2))) float v2f;
typedef __attribute__((ext_vector_type(8))) float v8f;
typedef unsigned long long ull;

#define BB 8
#define SS 1024
#define HH 512
#define EE 10
#define TOK (BB*SS)      // 8192 tokens
#define NSORT 16384      // padded sort size per batch row (>= SS*EE = 10240)

__device__ __forceinline__ v8f wmma4(v2f a, v2f b, v8f c) {
  // V_WMMA_F32_16X16X4_F32 : D = A(16x4) * B(4x16) + C(16x16), fp32 throughout
  return __builtin_amdgcn_wmma_f32_16x16x4_f32(false, a, false, b, (short)0, c, false, false);
}

__device__ __forceinline__ float sigmoidf_(float x) { return 1.f / (1.f + __expf(-x)); }

// ---------------- router ----------------
__global__ void router_logits(const float* __restrict__ x, const float* __restrict__ rw,
                              float* __restrict__ logits) {
  int gid = blockIdx.x * blockDim.x + threadIdx.x;
  if (gid >= TOK * EE) return;
  int tok = gid / EE, e = gid % EE;
  const float* xr = x + (size_t)tok * HH;
  const float* wr = rw + e * HH;
  float acc = 0.f;
  for (int i = 0; i < HH; ++i) acc = fmaf(xr[i], wr[i], acc);
  logits[gid] = acc;    // layout (tok, e) == flattened (S,E) per batch row
}

__global__ void softmax10(const float* __restrict__ logits, float* __restrict__ probs) {
  int tok = blockIdx.x * blockDim.x + threadIdx.x;
  if (tok >= TOK) return;
  float l[EE]; float m = -1e30f;
  for (int e = 0; e < EE; ++e) { l[e] = logits[tok*EE + e]; m = fmaxf(m, l[e]); }
  float s = 0.f;
  for (int e = 0; e < EE; ++e) { l[e] = expf(l[e] - m); s += l[e]; }
  float inv = 1.f / s;
  for (int e = 0; e < EE; ++e) probs[tok*EE + e] = l[e] * inv;
}

// ---------------- NMS: keys, bitonic sort, serial claim ----------------
__global__ void build_keys(const float* __restrict__ logits, ull* __restrict__ keys) {
  int gid = blockIdx.x * blockDim.x + threadIdx.x;
  if (gid >= BB * NSORT) return;
  int b = gid >> 14, i = gid & (NSORT - 1);
  ull key;
  if (i < SS * EE) {
    float f = logits[b * SS * EE + i];
    unsigned u = __float_as_uint(f);
    u = (u & 0x80000000u) ? ~u : (u | 0x80000000u); // ascending-orderable
    unsigned kd = ~u;                               // descending score
    key = ((ull)kd << 32) | (unsigned)i;            // ties: ascending index (stable)
  } else {
    key = ~0ull;                                    // padding, sorts to end
  }
  keys[gid] = key;
}

__global__ void bitonic_sort(ull* __restrict__ keys) {
  ull* kp = keys + (size_t)blockIdx.x * NSORT;
  int tid = threadIdx.x;
  for (unsigned kk = 2; kk <= NSORT; kk <<= 1) {
    for (unsigned j = kk >> 1; j > 0; j >>= 1) {
      __syncthreads();
      for (unsigned i = tid; i < NSORT; i += 1024) {
        unsigned ixj = i ^ j;
        if (ixj > i) {
          ull a = kp[i], bv = kp[ixj];
          bool up = ((i & kk) == 0);
          if ((a > bv) == up) { kp[i] = bv; kp[ixj] = a; }
        }
      }
    }
  }
}

__global__ void nms_kernel(const ull* __restrict__ keys, int* __restrict__ mask) {
  int b = blockIdx.x, tid = threadIdx.x;
  mask[b * SS + tid] = 0;
  __shared__ ull chunk[1024];
  unsigned occ[32];
  if (tid == 0) for (int i = 0; i < 32; ++i) occ[i] = 0u;
  const ull* kp = keys + (size_t)b * NSORT;
  for (int ch = 0; ch < NSORT / 1024; ++ch) {
    __syncthreads();
    chunk[tid] = kp[ch * 1024 + tid];
    __syncthreads();
    if (tid == 0) {
      for (int ii = 0; ii < 1024; ++ii) {
        unsigned idx = (unsigned)chunk[ii];
        if (idx >= SS * EE) continue;
        int pos = idx / EE, k = (int)(idx % EE) + 1;
        int st = pos - k / 2; if (st < 0) st = 0;
        int en = pos + (k + 1) / 2; if (en > SS) en = SS;
        int w0 = st >> 5, w1 = (en - 1) >> 5;
        unsigned m0b, m1b;
        if (w0 == w1) {
          m0b = ((1u << (en - st)) - 1u) << (st & 31);
          m1b = 0u;
        } else {
          m0b = 0xffffffffu << (st & 31);
          unsigned eb = (unsigned)en & 31u;
          m1b = eb ? ((1u << eb) - 1u) : 0xffffffffu;
        }
        bool freeW = ((occ[w0] & m0b) == 0u) && ((w0 == w1) || ((occ[w1] & m1b) == 0u));
        if (freeW) {
          occ[w0] |= m0b;
          if (w1 != w0) occ[w1] |= m1b;
          mask[b * SS + pos] = k;
        }
      }
    }
  }
}

// ---------------- weight transposes (K-pair packed: float2 = (B[2p][n], B[2p+1][n])) ----------------
__global__ void transpose_win(const float* __restrict__ w, v2f* __restrict__ wP) {
  int i = blockIdx.x * blockDim.x + threadIdx.x;       // over (HH/2) * 2HH pairs
  if (i >= (HH/2) * 2 * HH) return;
  int p = i / (2 * HH), j = i % (2 * HH);
  wP[i] = *(const v2f*)(w + j * HH + 2 * p);           // (w_in[j][2p], w_in[j][2p+1])
}
__global__ void transpose_wout(const float* __restrict__ w, v2f* __restrict__ wP) {
  int i = blockIdx.x * blockDim.x + threadIdx.x;       // over (HH/2) * HH pairs
  if (i >= (HH/2) * HH) return;
  int p = i / HH, o = i % HH;
  wP[i] = *(const v2f*)(w + o * HH + 2 * p);
}
__global__ void transpose_conv(const float* __restrict__ w, v2f* __restrict__ wP, int K, int HG) {
  int i = blockIdx.x * blockDim.x + threadIdx.x;       // over K*(HG/2)*HH pairs
  if (i >= K * (HG/2) * HH) return;
  int o = i % HH; int r = i / HH; int cp = r % (HG/2); int t = r / (HG/2);
  v2f v;
  v.x = w[(o * HG + 2*cp    ) * K + t];
  v.y = w[(o * HG + 2*cp + 1) * K + t];
  wP[i] = v;                                           // wP[t][cp][o]
}

// ---------------- GEMM1: gi = x @ w_in^T, GLU -> y (16 tok x 64 col per wave) ----------------
__global__ __launch_bounds__(32) void gemm1_glu(const float* __restrict__ x,
    const v2f* __restrict__ wP, const int* __restrict__ mask,
    float* __restrict__ y, int K) {
  int m0 = blockIdx.x * 16;        // flat token tile
  int n0 = blockIdx.y * 64;        // y-column tile
  int lane = threadIdx.x;
  int b = m0 / SS, s0 = m0 % SS;
  // halo must cover full receptive field of any conv tile containing a match
  int M = K + 16;
  bool hit = false;
  for (int s = s0 - M + lane; s < s0 + 16 + M; s += 32)
    if (s >= 0 && s < SS && mask[b * SS + s] == K) hit = true;
  if (__ballot(hit ? 1 : 0) == 0ull) return;

  v8f ag0={},ag1={},ag2={},ag3={}, ai0={},ai1={},ai2={},ai3={};
  int mr = m0 + (lane & 15);
  int ph = lane >> 4;              // pair half: K {0,1} vs {2,3}
  int nc = lane & 15;
  const float* xr = x + (size_t)mr * HH;
  for (int k0 = 0; k0 < HH; k0 += 4) {
    v2f a = *(const v2f*)(xr + k0 + 2 * ph);
    const v2f* wrow = wP + (size_t)((k0 >> 1) + ph) * (2 * HH);
    v2f bg0 = wrow[n0 + nc],          bg1 = wrow[n0 + 16 + nc];
    v2f bg2 = wrow[n0 + 32 + nc],     bg3 = wrow[n0 + 48 + nc];
    v2f bi0 = wrow[HH + n0 + nc],     bi1 = wrow[HH + n0 + 16 + nc];
    v2f bi2 = wrow[HH + n0 + 32 + nc],bi3 = wrow[HH + n0 + 48 + nc];
    ag0 = wmma4(a, bg0, ag0);
    ag1 = wmma4(a, bg1, ag1);
    ag2 = wmma4(a, bg2, ag2);
    ag3 = wmma4(a, bg3, ag3);
    ai0 = wmma4(a, bi0, ai0);
    ai1 = wmma4(a, bi1, ai1);
    ai2 = wmma4(a, bi2, ai2);
    ai3 = wmma4(a, bi3, ai3);
  }
  int rbase = m0 + 8 * (lane >> 4);
#pragma unroll
  for (int v = 0; v < 8; ++v) {
    size_t o = (size_t)(rbase + v) * HH + n0 + nc;
    y[o]      = ai0[v] * sigmoidf_(ag0[v]);
    y[o + 16] = ai1[v] * sigmoidf_(ag1[v]);
    y[o + 32] = ai2[v] * sigmoidf_(ag2[v]);
    y[o + 48] = ai3[v] * sigmoidf_(ag3[v]);
  }
}

// ---------------- grouped conv1d as k shifted grouped GEMMs ----------------
__global__ __launch_bounds__(32) void conv_grouped(const float* __restrict__ y,
    const v2f* __restrict__ cP, const float* __restrict__ cb,
    const int* __restrict__ mask, float* __restrict__ cbuf, int K, int HG) {
  int m0 = blockIdx.x * 16;
  int n0 = blockIdx.y * 64;        // 64 out-channels, always inside one group (HG>=64)
  int lane = threadIdx.x;
  bool hit = (lane < 16) && (mask[m0 + lane] == K);
  if (__ballot(hit ? 1 : 0) == 0ull) return;
  int b = m0 / SS, s0 = m0 % SS;
  int cibase = (n0 / HG) * HG;
  int ph = lane >> 4;
  int nc = lane & 15;
  int HG2 = HG >> 1;
  v8f a0 = {}, a1 = {}, a2 = {}, a3 = {};
  for (int t = 0; t < K; ++t) {
    int srow = s0 + (lane & 15) + t - K / 2;
    bool inb = (srow >= 0 && srow < SS);
    const float* yr = y + ((size_t)(b * SS) + srow) * HH + cibase;
    const v2f* wtb = cP + (size_t)t * HG2 * HH;
    for (int ci = 0; ci < HG; ci += 4) {
      int cc = ci + 2 * ph;
      float ax = 0.f, ay = 0.f;
      if (inb) { ax = yr[cc]; ay = yr[cc + 1]; }
      v2f a; a.x = ax; a.y = ay;
      const v2f* bp = wtb + (size_t)((ci >> 1) + ph) * HH + n0 + nc;
      v2f b0 = bp[0], b1 = bp[16], b2 = bp[32], b3 = bp[48];
      a0 = wmma4(a, b0, a0);
      a1 = wmma4(a, b1, a1);
      a2 = wmma4(a, b2, a2);
      a3 = wmma4(a, b3, a3);
    }
  }
  int rbase = m0 + 8 * (lane >> 4);
#pragma unroll
  for (int v = 0; v < 8; ++v) {
    size_t o = (size_t)(rbase + v) * HH + n0 + nc;
    cbuf[o]      = a0[v] + cb[n0 + nc];
    cbuf[o + 16] = a1[v] + cb[n0 + 16 + nc];
    cbuf[o + 32] = a2[v] + cb[n0 + 32 + nc];
    cbuf[o + 48] = a3[v] + cb[n0 + 48 + nc];
  }
}

// ---------------- LayerNorm + SiLU (only where mask==K), in place ----------------
__global__ void ln_silu(float* __restrict__ cbuf, const float* __restrict__ lnw,
                        const float* __restrict__ lnb, const int* __restrict__ mask, int K) {
  int tok = blockIdx.x * 8 + (threadIdx.x >> 5);
  int lane = threadIdx.x & 31;
  if (mask[tok] != K) return;
  float* row = cbuf + (size_t)tok * HH;
  float vals[16]; float s = 0.f, s2 = 0.f;
#pragma unroll
  for (int i = 0; i < 16; ++i) { float v = row[lane + i * 32]; vals[i] = v; s += v; s2 += v * v; }
  for (int o = 16; o >= 1; o >>= 1) { s += __shfl_xor(s, o, 32); s2 += __shfl_xor(s2, o, 32); }
  float mean = s * (1.f / HH);
  float var  = s2 * (1.f / HH) - mean * mean;
  float inv  = rsqrtf(var + 1e-5f);
#pragma unroll
  for (int i = 0; i < 16; ++i) {
    int col = lane + i * 32;
    float n = (vals[i] - mean) * inv * lnw[col] + lnb[col];
    row[col] = n * sigmoidf_(n);
  }
}

// ---------------- GEMM2: out = n @ w_out^T, masked scale by probs (16 x 64 per wave) ----------------
__global__ __launch_bounds__(32) void gemm2_scatter(const float* __restrict__ nb,
    const v2f* __restrict__ wP, const float* __restrict__ probs,
    const int* __restrict__ mask, float* __restrict__ agg, int K) {
  int m0 = blockIdx.x * 16;
  int n0 = blockIdx.y * 64;
  int lane = threadIdx.x;
  bool hit = (lane < 16) && (mask[m0 + lane] == K);
  if (__ballot(hit ? 1 : 0) == 0ull) return;
  v8f a0 = {}, a1 = {}, a2 = {}, a3 = {};
  int mr = m0 + (lane & 15);
  int ph = lane >> 4;
  int nc = lane & 15;
  const float* ar = nb + (size_t)mr * HH;
  for (int k0 = 0; k0 < HH; k0 += 4) {
    v2f a = *(const v2f*)(ar + k0 + 2 * ph);
    const v2f* wrow = wP + (size_t)((k0 >> 1) + ph) * HH;
    v2f b0 = wrow[n0 + nc],      b1 = wrow[n0 + 16 + nc];
    v2f b2 = wrow[n0 + 32 + nc], b3 = wrow[n0 + 48 + nc];
    a0 = wmma4(a, b0, a0);
    a1 = wmma4(a, b1, a1);
    a2 = wmma4(a, b2, a2);
    a3 = wmma4(a, b3, a3);
  }
  int rbase = m0 + 8 * (lane >> 4);
#pragma unroll
  for (int v = 0; v < 8; ++v) {
    int tok = rbase + v;
    if (mask[tok] == K) {                      // tokens are claimed by exactly one expert
      float p = probs[tok * EE + (K - 1)];
      size_t o = (size_t)tok * HH + n0 + nc;
      agg[o]      = a0[v] * p;
      agg[o + 16] = a1[v] * p;
      agg[o + 32] = a2[v] * p;
      agg[o + 48] = a3[v] * p;
    }
  }
}

// ---------------- epilogue: loss, compaction ----------------
__global__ void zero_f32(float* __restrict__ p, int n) {
  int i = blockIdx.x * blockDim.x + threadIdx.x;
  if (i < n) p[i] = 0.f;
}

__global__ void loss_kernel(const int* __restrict__ mask, float* __restrict__ outp) {
  __shared__ int scnt[EE];
  int tid = threadIdx.x;
  if (tid < EE) scnt[tid] = 0;
  __syncthreads();
  for (int i = tid; i < TOK; i += 1024) { int m = mask[i]; if (m > 0) atomicAdd(&scnt[m - 1], 1); }
  __syncthreads();
  if (tid == 0) {
    float tot = 0.f, c[EE];
    for (int e = 0; e < EE; ++e) { c[e] = (float)scnt[e]; tot += c[e]; }
    float loss = 0.f;
    if (tot > 0.f) {
      float d = fmaxf(tot, 1.f), mean = 0.f, nrm[EE];
      for (int e = 0; e < EE; ++e) { nrm[e] = c[e] / d; mean += nrm[e]; }
      mean *= (1.f / EE);
      float v = 0.f;
      for (int e = 0; e < EE; ++e) { float t = nrm[e] - mean; v += t * t; }
      loss = (v / (EE - 1)) * 0.01f;
    }
    outp[(size_t)TOK * HH] = loss;
  }
}

__global__ void build_selmap(const int* __restrict__ mask, int* __restrict__ selmap,
                             int* __restrict__ nsel) {
  int b = blockIdx.x, tid = threadIdx.x;
  __shared__ int sc[SS];
  int flag = (mask[b * SS + tid] > 0) ? 1 : 0;
  sc[tid] = flag;
  __syncthreads();
  for (int off = 1; off < SS; off <<= 1) {
    int v = (tid >= off) ? sc[tid - off] : 0;
    __syncthreads();
    sc[tid] += v;
    __syncthreads();
  }
  if (flag) selmap[b * SS + (sc[tid] - 1)] = tid;
  if (tid == SS - 1) nsel[b] = sc[SS - 1];
}

__global__ void scatter_out(const float* __restrict__ agg, const int* __restrict__ selmap,
                            const int* __restrict__ nsel, float* __restrict__ outp) {
  int blk = blockIdx.x;              // 0..TOK-1
  int b = blk / SS, j = blk % SS;
  int tid = threadIdx.x;
  float* dst = outp + (size_t)blk * HH;
  if (j < nsel[b]) {
    const float* src = agg + ((size_t)(b * SS) + selmap[b * SS + j]) * HH;
    for (int h = tid; h < HH; h += blockDim.x) dst[h] = src[h];
  } else {
    for (int h = tid; h < HH; h += blockDim.x) dst[h] = 0.f;
  }
}

extern "C" void kernel_launch(void* const* d_in, const int* in_sizes, int n_in,
                              void* d_out, int out_size, void* d_ws, size_t ws_size,
                              hipStream_t stream) {
  (void)in_sizes; (void)n_in; (void)out_size; (void)ws_size;
  const float* x  = (const float*)d_in[0];
  const float* rw = (const float*)d_in[1];
  float* outp = (float*)d_out;

  char* ws = (char*)d_ws;
  size_t off = 0;
  auto alloc = [&](size_t bytes) -> void* {
    void* p = ws + off;
    off = (off + bytes + 255) & ~(size_t)255;
    return p;
  };
  float* logits = (float*)alloc((size_t)TOK * EE * 4);
  float* probs  = (float*)alloc((size_t)TOK * EE * 4);
  ull*   keys   = (ull*)  alloc((size_t)BB * NSORT * 8);
  int*   mask   = (int*)  alloc((size_t)TOK * 4);
  int*   selmap = (int*)  alloc((size_t)TOK * 4);
  int*   nsel   = (int*)  alloc(64);
  float* ybuf   = (float*)alloc((size_t)TOK * HH * 4);
  float* cbuf   = (float*)alloc((size_t)TOK * HH * 4);
  float* agg    = (float*)alloc((size_t)TOK * HH * 4);
  v2f*   winP   = (v2f*)  alloc((size_t)HH * 2 * HH * 4);   // (HH/2)*(2HH) float2
  v2f*   woutP  = (v2f*)  alloc((size_t)HH * HH * 4);       // (HH/2)*HH float2
  v2f*   convP  = (v2f*)  alloc((size_t)2 * 1024 * 1024);   // max K*(HG/2)*HH float2 = 1.84MB

  router_logits<<<dim3((TOK * EE + 255) / 256), dim3(256), 0, stream>>>(x, rw, logits);
  softmax10<<<dim3((TOK + 255) / 256), dim3(256), 0, stream>>>(logits, probs);
  build_keys<<<dim3((BB * NSORT + 255) / 256), dim3(256), 0, stream>>>(logits, keys);
  bitonic_sort<<<dim3(BB), dim3(1024), 0, stream>>>(keys);
  nms_kernel<<<dim3(BB), dim3(1024), 0, stream>>>(keys, mask);
  zero_f32<<<dim3((TOK * HH + 255) / 256), dim3(256), 0, stream>>>(agg, TOK * HH);

  for (int e = 0; e < EE; ++e) {
    int K = e + 1;
    int g = 1;
    for (int c = K; c >= 1; --c) if (HH % c == 0) { g = c; break; }
    int HG = HH / g;
    const float* w_in  = (const float*)d_in[2 + 6 * e + 0];
    const float* convw = (const float*)d_in[2 + 6 * e + 1];
    const float* convb = (const float*)d_in[2 + 6 * e + 2];
    const float* lnw   = (const float*)d_in[2 + 6 * e + 3];
    const float* lnb   = (const float*)d_in[2 + 6 * e + 4];
    const float* w_out = (const float*)d_in[2 + 6 * e + 5];

    transpose_win<<<dim3(((HH/2) * 2 * HH + 255) / 256), dim3(256), 0, stream>>>(w_in, winP);
    transpose_wout<<<dim3(((HH/2) * HH + 255) / 256), dim3(256), 0, stream>>>(w_out, woutP);
    int cte = K * (HG/2) * HH;
    transpose_conv<<<dim3((cte + 255) / 256), dim3(256), 0, stream>>>(convw, convP, K, HG);

    gemm1_glu<<<dim3(TOK / 16, HH / 64), dim3(32), 0, stream>>>(x, winP, mask, ybuf, K);
    conv_grouped<<<dim3(TOK / 16, HH / 64), dim3(32), 0, stream>>>(ybuf, convP, convb, mask, cbuf, K, HG);
    ln_silu<<<dim3(TOK / 8), dim3(256), 0, stream>>>(cbuf, lnw, lnb, mask, K);
    gemm2_scatter<<<dim3(TOK / 16, HH / 64), dim3(32), 0, stream>>>(cbuf, woutP, probs, mask, agg, K);
  }

  loss_kernel<<<dim3(1), dim3(1024), 0, stream>>>(mask, outp);
  build_selmap<<<dim3(BB), dim3(1024), 0, stream>>>(mask, selmap, nsel);
  scatter_out<<<dim3(TOK), dim3(128), 0, stream>>>(agg, selmap, nsel, outp);
}